// TimeAwareLightGCN_26989574488332
// MI455X (gfx1250) — compile-verified
//
#include <hip/hip_runtime.h>
#include <hip/hip_bf16.h>
#include <math.h>

// ---------------------------------------------------------------------------
// TimeAwareLightGCN fused kernel for gfx1250 (MI455X).
//   out[b,:] = c[b,:136] @ W[136,64]      (f32 WMMA 16x16x4, K=136 = 34 chunks)
//   c[b, n*17+0]    = P[b,n]
//   c[b, n*17+1+k]  = P[b,n]*cos((k+1)*angle_b)
//   c[b, n*17+9+k]  = P[b,n]*sin((k+1)*angle_b)
//   W row n*17+0 = a0[n], rows +1..8 = fourier_a[n], rows +9..16 = fourier_b[n]
// Memory-bound problem (~72MB @ 23.3TB/s ~= 3.1us); W table cached in LDS.
// ---------------------------------------------------------------------------

typedef __attribute__((ext_vector_type(2))) float v2f;
typedef __attribute__((ext_vector_type(4))) float v4f;
typedef __attribute__((ext_vector_type(8))) float v8f;

#define TAU_INV   5.0f
#define NCLUST    8
#define FK        8
#define DIM       64
#define CPC       17          // coeffs per cluster: 1 + 2*FK
#define KTOT      136         // NCLUST*CPC
#define NCHUNK    34          // KTOT/4
#define NMIN      1440
#define TWO_PI_F  6.283185307179586f
#define ITEMS_PER_WAVE  16
#define WAVES_PER_BLOCK 8
#define ITEMS_PER_BLOCK 128

__global__ __launch_bounds__(256) void talgcn_fused_kernel(
    const float* __restrict__ emb,      // (B,64)
    const float* __restrict__ thetas,   // (B,)
    const float* __restrict__ protos,   // (8,64)
    const float* __restrict__ a0,       // (8,64)
    const float* __restrict__ fa,       // (8,8,64)
    const float* __restrict__ fb,       // (8,8,64)
    float* __restrict__ out_emb,        // (B,64)
    float* __restrict__ out_P,          // (B,8)
    int B)
{
    __shared__ __align__(16) float Wl[KTOT * DIM];    // 34816 B weight table
    __shared__ __align__(16) float Pl[NCLUST * DIM];  //  2048 B prototypes

    const int tid = threadIdx.x;

    // ---- cooperative build of W table + prototype cache in LDS ----
    for (int idx = tid; idx < KTOT * DIM; idx += 256) {
        int row = idx >> 6;          // /DIM
        int d   = idx & (DIM - 1);
        int n   = row / CPC;
        int t   = row - n * CPC;
        float v;
        if (t == 0)       v = a0[n * DIM + d];
        else if (t <= FK) v = fa[(n * FK + (t - 1)) * DIM + d];
        else              v = fb[(n * FK + (t - 1 - FK)) * DIM + d];
        Wl[idx] = v;
    }
    for (int idx = tid; idx < NCLUST * DIM; idx += 256)
        Pl[idx] = protos[idx];
    __syncthreads();

    const int lane = tid & 31;
    const int wave = tid >> 5;
    const int half = lane >> 4;          // 0: lanes 0-15, 1: lanes 16-31
    const int mrow = lane & 15;          // M row / output column within tile
    const int base = blockIdx.x * ITEMS_PER_BLOCK + wave * ITEMS_PER_WAVE;
    if (base >= B) return;               // (B is a multiple of 128; safety only)
    const int item = base + mrow;        // both lane-halves own the same item

    // ---- per-item cluster scores -> softmax P ----
    float acc[NCLUST];
    #pragma unroll
    for (int n = 0; n < NCLUST; ++n) acc[n] = 0.0f;
    const v4f* e4 = (const v4f*)(emb + (size_t)item * DIM);
    #pragma unroll
    for (int d4 = 0; d4 < DIM / 4; ++d4) {
        v4f e = e4[d4];
        #pragma unroll
        for (int n = 0; n < NCLUST; ++n) {
            v4f p = *(const v4f*)&Pl[n * DIM + d4 * 4];
            acc[n] += e.x * p.x + e.y * p.y + e.z * p.z + e.w * p.w;
        }
    }
    float mx = acc[0] * TAU_INV;
    #pragma unroll
    for (int n = 1; n < NCLUST; ++n) mx = fmaxf(mx, acc[n] * TAU_INV);
    float Pv[NCLUST];
    float sum = 0.0f;
    #pragma unroll
    for (int n = 0; n < NCLUST; ++n) {
        Pv[n] = __expf(acc[n] * TAU_INV - mx);
        sum += Pv[n];
    }
    float rs = __frcp_rn(sum);
    #pragma unroll
    for (int n = 0; n < NCLUST; ++n) Pv[n] *= rs;

    // ---- minute bucket + fourier basis ----
    float th = thetas[item];
    int minute = (int)(th * ((float)NMIN / TWO_PI_F));
    minute = min(max(minute, 0), NMIN - 1);
    float ang = (float)minute * (TWO_PI_F / (float)(NMIN - 1));
    float cv[FK], sv[FK];
    #pragma unroll
    for (int k = 0; k < FK; ++k)
        __sincosf((float)(k + 1) * ang, &sv[k], &cv[k]);

    // coefficient c[j] for this lane's item (j must be compile-time constant)
    auto cval = [&](int j) -> float {
        int n = j / CPC;
        int t = j - n * CPC;
        if (t == 0)  return Pv[n];
        if (t <= FK) return Pv[n] * cv[t - 1];
        return Pv[n] * sv[t - 1 - FK];
    };

    // ---- mixture GEMM: (16 x 136) x (136 x 64) via V_WMMA_F32_16X16X4_F32 ----
    v8f dacc0 = {0.0f}, dacc1 = {0.0f}, dacc2 = {0.0f}, dacc3 = {0.0f};
    // B operand LDS column base: row-part (half*2) folded in, plus this lane's col
    const int bbase = half * 2 * DIM + mrow;

    #pragma unroll
    for (int ch = 0; ch < NCHUNK; ++ch) {
        const int j0 = ch * 4;
        // A-matrix (16x4 f32): lane<16 -> K=j0,j0+1 ; lane>=16 -> K=j0+2,j0+3
        float c0 = cval(j0 + 0), c1 = cval(j0 + 1);
        float c2 = cval(j0 + 2), c3 = cval(j0 + 3);
        v2f A;
        A.x = half ? c2 : c0;
        A.y = half ? c3 : c1;
        // B-matrix (4x16 f32) per N-tile, mirrored half-wave K split
        #pragma unroll
        for (int nt = 0; nt < 4; ++nt) {
            v2f Bm;
            Bm.x = Wl[(j0 + 0) * DIM + bbase + nt * 16];
            Bm.y = Wl[(j0 + 1) * DIM + bbase + nt * 16];
            v8f cc = (nt == 0) ? dacc0 : (nt == 1) ? dacc1 : (nt == 2) ? dacc2 : dacc3;
            cc = __builtin_amdgcn_wmma_f32_16x16x4_f32(
                false, A, false, Bm, (short)0, cc, false, false);
            if      (nt == 0) dacc0 = cc;
            else if (nt == 1) dacc1 = cc;
            else if (nt == 2) dacc2 = cc;
            else              dacc3 = cc;
        }
    }

    // ---- store D tiles: VGPR r -> row r + 8*half, col = mrow (+ nt*16) ----
    #pragma unroll
    for (int r = 0; r < 8; ++r) {
        int row = r + half * 8;
        size_t o = (size_t)(base + row) * DIM + mrow;
        out_emb[o +  0] = dacc0[r];
        out_emb[o + 16] = dacc1[r];
        out_emb[o + 32] = dacc2[r];
        out_emb[o + 48] = dacc3[r];
    }
    // ---- store P (one lane-half only, to avoid duplicate writes) ----
    if (lane < 16) {
        #pragma unroll
        for (int n = 0; n < NCLUST; ++n)
            out_P[(size_t)item * NCLUST + n] = Pv[n];
    }
}

extern "C" void kernel_launch(void* const* d_in, const int* in_sizes, int n_in,
                              void* d_out, int out_size, void* d_ws, size_t ws_size,
                              hipStream_t stream) {
    const float* emb    = (const float*)d_in[0];   // (B,64)
    const float* thetas = (const float*)d_in[1];   // (B,)
    const float* protos = (const float*)d_in[2];   // (8,64)
    const float* a0     = (const float*)d_in[3];   // (8,64)
    const float* fa     = (const float*)d_in[4];   // (8,8,64)
    const float* fb     = (const float*)d_in[5];   // (8,8,64)

    const int B = in_sizes[1];                     // 131072
    float* out_emb = (float*)d_out;                // (B,64)
    float* out_P   = out_emb + (size_t)B * DIM;    // (B,8)

    const int grid = (B + ITEMS_PER_BLOCK - 1) / ITEMS_PER_BLOCK;
    talgcn_fused_kernel<<<grid, 256, 0, stream>>>(
        emb, thetas, protos, a0, fa, fb, out_emb, out_P, B);
}